// MultiHeadAttention_77232101917222
// MI455X (gfx1250) — compile-verified
//
#include <hip/hip_runtime.h>
#include <math.h>
#include <stdint.h>

// ---------------- problem constants ----------------
#define D_MODEL   1024
#define NUM_HEADS 16
#define DEPTH     64
#define SEQ       2048
#define BATCH     2
#define M_TOTAL   (BATCH * SEQ)   // 4096 rows for all projection GEMMs
#define ROWS      16              // query rows per attention block
#define KC        32              // GEMM K-chunk staged per TDM transfer
#define LDA       34              // padded LDS row stride (32 data + 2 pad dwords)

typedef __attribute__((ext_vector_type(2))) float v2f;
typedef __attribute__((ext_vector_type(8))) float v8f;
typedef __attribute__((ext_vector_type(4))) unsigned int u32x4;
typedef __attribute__((ext_vector_type(8))) int i32x8;
typedef __attribute__((ext_vector_type(4))) int i32x4;

// CDNA5 native fp32 matrix op: D(16x16) = A(16x4) * B(4x16) + C
static __device__ __forceinline__ v8f wmma_f32(v2f a, v2f b, v8f c) {
  return __builtin_amdgcn_wmma_f32_16x16x4_f32(
      /*neg_a=*/false, a, /*neg_b=*/false, b,
      /*c_mod=*/(short)0, c, /*reuse_a=*/false, /*reuse_b=*/false);
}

// Low 32 bits of a flat shared-space pointer == LDS byte offset (ISA 10.2).
static __device__ __forceinline__ unsigned lds_off(const void* p) {
  return (unsigned)(unsigned long long)(uintptr_t)p;
}

// ---------------------------------------------------------------------------
// Tensor Data Mover: issue a 2D f32 tile load Global->LDS (ISA ch.8 D#).
//   tile_cols elements contiguous (dim0), tile_rows rows, row stride in
//   elements. pad=true inserts 2 dwords after every 32 dwords stored so each
//   32-float tile row lands at LDS stride 34 (bank-conflict-free fragments).
// ---------------------------------------------------------------------------
static __device__ __forceinline__ void tdm_load_tile_f32(
    const float* gptr, unsigned lds_byte_off,
    unsigned tile_cols, unsigned tile_rows,
    unsigned tensor_cols, unsigned tensor_rows,
    unsigned row_stride_elems, bool pad) {
  unsigned long long ga = (unsigned long long)(uintptr_t)gptr;
  u32x4 g0;
  g0.x = 1u;                                       // count=1, user descriptor
  g0.y = lds_byte_off;                             // lds_addr
  g0.z = (unsigned)ga;                             // global_addr[31:0]
  g0.w = (unsigned)((ga >> 32) & 0x01ffffffu) | (2u << 30);  // addr[56:32]|type=2
  i32x8 g1;
  unsigned w0 = (2u << 16);                        // data_size = 4 bytes
  if (pad) w0 |= (1u << 20)                        // pad_enable
             | (4u << 22)                          // pad_interval: every 32 dwords
             | (1u << 25);                         // pad_amount: 2 dwords
  g1[0] = (int)w0;                                 // workgroup_mask=0 (no cluster)
  g1[1] = (int)((tensor_cols & 0xffffu) << 16);    // tensor_dim0[15:0]
  g1[2] = (int)((tensor_cols >> 16) | ((tensor_rows & 0xffffu) << 16));
  g1[3] = (int)((tensor_rows >> 16) | ((tile_cols & 0xffffu) << 16)); // tile_dim0
  g1[4] = (int)(tile_rows & 0xffffu);              // tile_dim1 (tile_dim2=0)
  g1[5] = (int)row_stride_elems;                   // tensor_dim0_stride[31:0]
  g1[6] = 0;
  g1[7] = 0;
  i32x4 zz4 = {0, 0, 0, 0};                        // 2D: groups 2/3 unused
  i32x8 zz8 = {0, 0, 0, 0, 0, 0, 0, 0};
  __builtin_amdgcn_tensor_load_to_lds(g0, g1, zz4, zz4, zz8, 0);
}

// ---------------------------------------------------------------------------
// Projection GEMM:  Y = X @ W^T + bias   (X:[4096,1024], W:[1024,1024])
//   TDM double-buffered LDS staging, f32 WMMA compute.
//   Block 256 threads = 8 waves, block tile 64(M) x 128(N), K-chunk 32.
//   mode 0: write split-head [B,H,S,DEPTH]; mode 1: write flat [M,1024].
// ---------------------------------------------------------------------------
__global__ __launch_bounds__(256)
void proj_gemm_kernel(const float* __restrict__ X, const float* __restrict__ W,
                      const float* __restrict__ bias, float* __restrict__ Y,
                      int mode) {
  __shared__ float sA[2][64 * LDA];    // 2 x 8.5 KB  (X tile, padded rows)
  __shared__ float sB[2][128 * LDA];   // 2 x 17 KB   (W tile, padded rows)

  const int lane = threadIdx.x & 31;
  const int wv   = threadIdx.x >> 5;          // wave id 0..7
  const int hl   = lane >> 4;                 // half-wave (0/1)
  const int ll   = lane & 15;
  const int mblk = blockIdx.x * 64;
  const int nblk = blockIdx.y * 128;

  if (wv == 0) {                              // prime chunk 0
    tdm_load_tile_f32(X + (size_t)mblk * D_MODEL, lds_off(&sA[0][0]),
                      KC, 64, D_MODEL, M_TOTAL, D_MODEL, true);
    tdm_load_tile_f32(W + (size_t)nblk * D_MODEL, lds_off(&sB[0][0]),
                      KC, 128, D_MODEL, D_MODEL, D_MODEL, true);
  }

  v8f acc[4] = {};
  const int NCH = D_MODEL / KC;               // 32 chunks
  for (int t = 0; t < NCH; ++t) {
    const int cur = t & 1;
    if (wv == 0) {
      if (t + 1 < NCH) {                      // issue chunk t+1 into other buffer
        tdm_load_tile_f32(X + (size_t)mblk * D_MODEL + (t + 1) * KC,
                          lds_off(&sA[cur ^ 1][0]),
                          KC, 64, D_MODEL, M_TOTAL, D_MODEL, true);
        tdm_load_tile_f32(W + (size_t)nblk * D_MODEL + (t + 1) * KC,
                          lds_off(&sB[cur ^ 1][0]),
                          KC, 128, D_MODEL, D_MODEL, D_MODEL, true);
        __builtin_amdgcn_s_wait_tensorcnt(2); // chunk t's two DMAs are done
      } else {
        __builtin_amdgcn_s_wait_tensorcnt(0);
      }
    }
    __syncthreads();                          // publish buffer `cur`

    const float* As = &sA[cur][((wv & 3) * 16 + ll) * LDA];
    const float* Bs = &sB[cur][((wv >> 2) * 64 + ll) * LDA];
#pragma unroll
    for (int ks = 0; ks < KC; ks += 4) {
      const int kk = ks + 2 * hl;             // lane holds f32 pair (k, k+1)
      v2f a = *(const v2f*)(As + kk);
#pragma unroll
      for (int t4 = 0; t4 < 4; ++t4) {
        v2f b = *(const v2f*)(Bs + t4 * 16 * LDA + kk);
        acc[t4] = wmma_f32(a, b, acc[t4]);
      }
    }
    __syncthreads();                          // done reading; buffer reusable
  }

  // ---- epilogue: bias + store
  const int m0 = mblk + (wv & 3) * 16;
  const int nb = nblk + (wv >> 2) * 64;
#pragma unroll
  for (int t4 = 0; t4 < 4; ++t4) {
    const int col = nb + t4 * 16 + ll;
    const float bv = bias[col];
#pragma unroll
    for (int i = 0; i < 8; ++i) {
      const int r = m0 + i + 8 * hl;          // C layout: lanes>=16 hold M=i+8
      const float val = acc[t4][i] + bv;
      if (mode == 0) {                        // split heads: [B,H,S,DEPTH]
        const int b_ = r >> 11, s = r & (SEQ - 1);
        const int h = col >> 6, d = col & (DEPTH - 1);
        Y[(((size_t)(b_ * NUM_HEADS + h) * SEQ) + s) * DEPTH + d] = val;
      } else {                                // flat [M_TOTAL, D_MODEL]
        Y[(size_t)r * D_MODEL + col] = val;
      }
    }
  }
}

// ---------------------------------------------------------------------------
// Fused attention per (b,h): 16 query rows per block.
//   phase 1: logits = scale * Q K^T  (f32 WMMA, batched fragment loads)
//   phase 2: row softmax in LDS; normalized P written to global attn output
//   phase 3: out = P @ V (A from LDS), split-K waves combined via ds_add_f32
// ---------------------------------------------------------------------------
__global__ __launch_bounds__(256)
void attention_kernel(const float* __restrict__ Q, const float* __restrict__ K,
                      const float* __restrict__ V, float* __restrict__ attn,
                      float* __restrict__ OH) {
  __shared__ float sP[ROWS * SEQ];     // 131072 B logits/probs slab
  __shared__ float sO[ROWS * DEPTH];   // 4096 B   P@V accumulator

  const int lane = threadIdx.x & 31;
  const int wv   = threadIdx.x >> 5;   // 0..7
  const int hl   = lane >> 4;
  const int ll   = lane & 15;
  const int bh   = blockIdx.y;                       // b*H + h
  const int q0   = blockIdx.x * ROWS;
  const size_t base = (size_t)bh * SEQ * DEPTH;      // head base in [B,H,S,d]

  for (int i = threadIdx.x; i < ROWS * DEPTH; i += 256) sO[i] = 0.f;

  // Preload all 16 A-fragments of the Q slab (DEPTH=64 -> 16 K-steps of 4).
  v2f aq[16];
  {
    const float* qrow = Q + base + (size_t)(q0 + ll) * DEPTH;
#pragma unroll
    for (int s = 0; s < 16; ++s)
      aq[s] = *(const v2f*)(qrow + 4 * s + 2 * hl);
  }

  // ---- phase 1: logits into LDS. Wave wv covers key cols [wv*256, wv*256+255].
  for (int j = 0; j < 16; ++j) {
    const int n0 = (wv * 16 + j) * 16;
    const float* krow = K + base + (size_t)(n0 + ll) * DEPTH; // B[k'][n]=K[n][k']
    v2f bf[16];
#pragma unroll
    for (int s = 0; s < 16; ++s)                 // batch loads: 16 in flight
      bf[s] = *(const v2f*)(krow + 4 * s + 2 * hl);
    v8f acc = {};
#pragma unroll
    for (int s = 0; s < 16; ++s)                 // then WMMA burst
      acc = wmma_f32(aq[s], bf[s], acc);
#pragma unroll
    for (int i = 0; i < 8; ++i) {
      const int r = i + 8 * hl;
      sP[r * SEQ + n0 + ll] = acc[i] * 0.125f;   // 1/sqrt(64)
    }
  }
  __syncthreads();

  // ---- phase 2: softmax, two rows per wave; write attn to global.
  for (int rr = 0; rr < 2; ++rr) {
    const int r = 2 * wv + rr;
    float* row = sP + r * SEQ;
    float m = -INFINITY;
    for (int c = lane; c < SEQ; c += 32) m = fmaxf(m, row[c]);
    for (int off = 16; off; off >>= 1) m = fmaxf(m, __shfl_xor(m, off, 32));
    float sum = 0.f;
    for (int c = lane; c < SEQ; c += 32) {
      const float e = __expf(row[c] - m);
      row[c] = e;
      sum += e;
    }
    for (int off = 16; off; off >>= 1) sum += __shfl_xor(sum, off, 32);
    const float inv = 1.f / sum;
    float* g = attn + ((size_t)bh * SEQ + (q0 + r)) * SEQ;
    for (int c = lane; c < SEQ; c += 32) {
      const float p = row[c] * inv;
      row[c] = p;                                // keep for phase 3
      g[c]   = p;                                // coalesced global store
    }
  }
  __syncthreads();

  // ---- phase 3: out(16x64) = P(16x2048) @ V(2048x64), split-K over wave pairs.
  {
    const int tc = wv & 3;            // depth tile 0..3
    const int kh = wv >> 2;           // K half 0/1
    const int n  = tc * 16 + ll;      // depth column
    v8f acc = {};
    const int kbeg = kh * (SEQ / 2);
    for (int k = kbeg; k < kbeg + SEQ / 2; k += 32) {   // 8 WMMA per batch
      v2f av[8], bv8[8];
#pragma unroll
      for (int u = 0; u < 8; ++u) {
        const int kk = k + 4 * u + 2 * hl;
        av[u] = *(const v2f*)(sP + ll * SEQ + kk);      // A from LDS
        bv8[u].x = V[base + (size_t)kk * DEPTH + n];    // B[k'][n] = V[k'][n]
        bv8[u].y = V[base + (size_t)(kk + 1) * DEPTH + n];
      }
#pragma unroll
      for (int u = 0; u < 8; ++u)
        acc = wmma_f32(av[u], bv8[u], acc);
    }
#pragma unroll
    for (int i = 0; i < 8; ++i) {
      const int r = i + 8 * hl;
      atomicAdd(&sO[r * DEPTH + n], acc[i]);            // ds_add_f32 combine
    }
  }
  __syncthreads();

  // ---- write concat-head layout [B, S, D_MODEL] into workspace.
  {
    const int b_ = bh >> 4, h = bh & 15;
    for (int idx = threadIdx.x; idx < ROWS * DEPTH; idx += 256) {
      const int r = idx >> 6, c = idx & 63;
      const int s = q0 + r;
      OH[((size_t)(b_ * SEQ + s)) * D_MODEL + h * DEPTH + c] = sO[idx];
    }
  }
}

// ---------------------------------------------------------------------------
extern "C" void kernel_launch(void* const* d_in, const int* in_sizes, int n_in,
                              void* d_out, int out_size, void* d_ws, size_t ws_size,
                              hipStream_t stream) {
  const float* query   = (const float*)d_in[0];
  const float* key     = (const float*)d_in[1];
  const float* value   = (const float*)d_in[2];
  const float* wq_w    = (const float*)d_in[3];
  const float* wq_b    = (const float*)d_in[4];
  const float* wk_w    = (const float*)d_in[5];
  const float* wk_b    = (const float*)d_in[6];
  const float* wv_w    = (const float*)d_in[7];
  const float* wv_b    = (const float*)d_in[8];
  const float* dense_w = (const float*)d_in[9];
  const float* dense_b = (const float*)d_in[10];

  float* out  = (float*)d_out;                           // [B,S,D]  4,194,304 f32
  float* attn = out + (size_t)M_TOTAL * D_MODEL;         // [B,H,S,S] 134,217,728 f32

  const size_t PROJ = (size_t)M_TOTAL * D_MODEL;         // 4,194,304 f32 each
  float* q  = (float*)d_ws;
  float* k  = q + PROJ;
  float* v  = k + PROJ;
  float* oh = v + PROJ;                                  // total 64 MB workspace

  dim3 blk(256);
  dim3 gproj(M_TOTAL / 64, D_MODEL / 128);               // 64 x 8 blocks
  proj_gemm_kernel<<<gproj, blk, 0, stream>>>(query, wq_w, wq_b, q, 0);
  proj_gemm_kernel<<<gproj, blk, 0, stream>>>(key,   wk_w, wk_b, k, 0);
  proj_gemm_kernel<<<gproj, blk, 0, stream>>>(value, wv_w, wv_b, v, 0);

  dim3 gattn(SEQ / ROWS, BATCH * NUM_HEADS);             // 128 x 32 blocks
  attention_kernel<<<gattn, blk, 0, stream>>>(q, k, v, attn, oh);

  proj_gemm_kernel<<<gproj, blk, 0, stream>>>(oh, dense_w, dense_b, out, 1);
}